// ClebschGordanDecomposer_3977139716373
// MI455X (gfx1250) — compile-verified
//
#include <hip/hip_runtime.h>
#include <hip/hip_bf16.h>
#include <stdint.h>

// Sparse E8 Lie bracket on MI455X (gfx1250, wave32).
//
// Roofline: ~260 MB HBM traffic/call -> ~11 us floor @ 23.3 TB/s.  Sparse
// compute (B*NNZ = 268M gather-mul-scatter, ~0.8 GFLOP) is on-chip LDS work.
// A dense/WMMA reformulation (contrib[B,NNZ] x scatter[NNZ,D]) would be
// 133 GFLOP dense -- a 165x blow-up -- so WMMA is the wrong tool here.
// The CDNA5-specific win is the async DMA path: COO coefficients and v1/v2 rows are
// staged global->LDS with GLOBAL_LOAD_ASYNC_TO_LDS_B128 (ASYNCcnt), antisym
// is drained LDS->global with GLOBAL_STORE_ASYNC_FROM_LDS_B128, and the
// scatter itself is single-instruction no-return DS_ADD_F32 atomics.

#define B_TOT   65536
#define D_DIM   248
#define D_VEC4  62                  // 248 floats = 62 x b128
#define ROW_BYTES (D_DIM * 4)       // 992 B, multiple of 16
#define NNZ     4096
#define THREADS 256
#define WAVES   8                   // wave32: 256 threads = 8 waves
#define ROWBUF  256                 // padded row buffer (248 used)
#define BLOCKS  1024
#define SLOTS   (BLOCKS * WAVES)    // 8192 concurrent row slots
#define ROW_ITERS (B_TOT / SLOTS)   // 8 rows per wave

__device__ __forceinline__ unsigned lds_addr_of(const void* p) {
    // Generic (flat) LDS pointers carry the wave-relative LDS byte offset in
    // their low 32 bits (flat aperture encoding) -> truncate.
    return (unsigned)(uintptr_t)p;
}

__device__ __forceinline__ void async_load_b128(unsigned lds, int goff, uint64_t gbase) {
    asm volatile("global_load_async_to_lds_b128 %0, %1, %2"
                 :: "v"(lds), "v"(goff), "s"(gbase) : "memory");
}

__device__ __forceinline__ void async_store_b128(int goff, unsigned lds, uint64_t gbase) {
    asm volatile("global_store_async_from_lds_b128 %0, %1, %2"
                 :: "v"(goff), "v"(lds), "s"(gbase) : "memory");
}

__device__ __forceinline__ void lds_add_f32(unsigned lds, float v) {
    asm volatile("ds_add_f32 %0, %1" :: "v"(lds), "v"(v) : "memory");
}

__device__ __forceinline__ void wait_async0() {
    asm volatile("s_wait_asynccnt 0" ::: "memory");
}

__device__ __forceinline__ void wait_ds0() {
    asm volatile("s_wait_dscnt 0" ::: "memory");
}

__global__ __launch_bounds__(THREADS)
void e8_bracket_kernel(const float* __restrict__ v1,
                       const float* __restrict__ v2,
                       const int*   __restrict__ Ig,
                       const int*   __restrict__ Jg,
                       const int*   __restrict__ Kg,
                       const float* __restrict__ Cg,
                       float* __restrict__ out_anti,
                       float* __restrict__ out_sym,
                       float* __restrict__ out_scalar)
{
    // 16 KB packed indices + 16 KB coefficients + 24 KB row buffers = 56 KB
    __shared__ __align__(16) unsigned s_pack[NNZ];             // I | J<<8 | K<<16
    __shared__ __align__(16) float    s_c[NNZ];
    __shared__ __align__(16) float    s_buf[WAVES][3][ROWBUF]; // v1row, v2row, acc

    const int tid  = threadIdx.x;
    const int wave = tid >> 5;
    const int lane = tid & 31;

    const uint64_t v1b   = (uint64_t)(uintptr_t)v1;
    const uint64_t v2b   = (uint64_t)(uintptr_t)v2;
    const uint64_t antib = (uint64_t)(uintptr_t)out_anti;

    // ---------------- Stage COO table into LDS ----------------
    {
        const unsigned lds_c = lds_addr_of(&s_c[0]);
        const uint64_t cbase = (uint64_t)(uintptr_t)Cg;
        for (int t = tid; t < NNZ / 4; t += THREADS)
            async_load_b128(lds_c + (unsigned)t * 16u, t * 16, cbase);
        // I/J/K packed into one dword per entry (all indices < 248 < 256).
        for (int t = tid; t < NNZ; t += THREADS)
            s_pack[t] = (unsigned)Ig[t] | ((unsigned)Jg[t] << 8)
                                        | ((unsigned)Kg[t] << 16);
        wait_async0();   // each wave drains its own DMA before the barrier
    }
    __syncthreads();

    float* v1r = s_buf[wave][0];
    float* v2r = s_buf[wave][1];
    float* acc = s_buf[wave][2];
    const unsigned v1r_lds = lds_addr_of(v1r);
    const unsigned v2r_lds = lds_addr_of(v2r);
    const unsigned acc_lds = lds_addr_of(acc);
    const uint4*  p4   = (const uint4*) s_pack;
    const float4* c4   = (const float4*)s_c;
    const float4* v1r4 = (const float4*)v1r;
    const float4* v2r4 = (const float4*)v2r;
    float4*       acc4 = (float4*)acc;

    const int slot = blockIdx.x * WAVES + wave;

    for (int it = 0; it < ROW_ITERS; ++it) {
        const int row   = slot + it * SLOTS;
        const int base4 = row * ROW_BYTES;          // byte offset, < 2^31
        const size_t basef = (size_t)row * D_DIM;   // float offset

        // Stage both rows via async DMA; zero the accumulator meanwhile.
        for (int t = lane; t < D_VEC4; t += 32) {
            async_load_b128(v1r_lds + (unsigned)t * 16u, base4 + t * 16, v1b);
            async_load_b128(v2r_lds + (unsigned)t * 16u, base4 + t * 16, v2b);
            acc4[t] = make_float4(0.f, 0.f, 0.f, 0.f);
        }
        wait_async0();   // rows resident in LDS (DS zeroing is in-order anyway)

        // Gather-multiply-scatter over the COO table, 4 entries/lane/iter.
        for (int t = lane; t < NNZ / 4; t += 32) {
            uint4  p = p4[t];
            float4 c = c4[t];
            {
                unsigned e = p.x;
                float v = v1r[e & 255u] * v2r[(e >> 8) & 255u] * c.x;
                lds_add_f32(acc_lds + ((e >> 14) & 0x3FCu), v);
            }
            {
                unsigned e = p.y;
                float v = v1r[e & 255u] * v2r[(e >> 8) & 255u] * c.y;
                lds_add_f32(acc_lds + ((e >> 14) & 0x3FCu), v);
            }
            {
                unsigned e = p.z;
                float v = v1r[e & 255u] * v2r[(e >> 8) & 255u] * c.z;
                lds_add_f32(acc_lds + ((e >> 14) & 0x3FCu), v);
            }
            {
                unsigned e = p.w;
                float v = v1r[e & 255u] * v2r[(e >> 8) & 255u] * c.w;
                lds_add_f32(acc_lds + ((e >> 14) & 0x3FCu), v);
            }
        }
        wait_ds0();      // all lane atomics landed in acc

        // antisym: DMA straight out of LDS, no VGPR round-trip.
        // sym + scalar: vectorized from the staged rows.
        float s = 0.0f;
        for (int t = lane; t < D_VEC4; t += 32) {
            async_store_b128(base4 + t * 16, acc_lds + (unsigned)t * 16u, antib);
            float4 a = v1r4[t];
            float4 b = v2r4[t];
            float4 pr = make_float4(a.x * b.x, a.y * b.y, a.z * b.z, a.w * b.w);
            *(float4*)(out_sym + basef + (size_t)t * 4) = pr;
            s += pr.x + pr.y + pr.z + pr.w;
        }

        // Wave32 reduction via the (now consumed) v2 row buffer.
        v2r[lane] = s;
        wait_ds0();
        if (lane == 0) {
            float tot = 0.0f;
            #pragma unroll
            for (int l = 0; l < 32; ++l) tot += v2r[l];
            out_scalar[row] = tot;
        }
        wait_ds0();      // reduction reads done before rows are overwritten
        wait_async0();   // acc DMA-out drained before re-zeroing next row
    }
}

extern "C" void kernel_launch(void* const* d_in, const int* in_sizes, int n_in,
                              void* d_out, int out_size, void* d_ws, size_t ws_size,
                              hipStream_t stream) {
    (void)in_sizes; (void)n_in; (void)out_size; (void)d_ws; (void)ws_size;
    const float* v1 = (const float*)d_in[0];
    const float* v2 = (const float*)d_in[1];
    const int*   I  = (const int*)  d_in[2];
    const int*   J  = (const int*)  d_in[3];
    const int*   K  = (const int*)  d_in[4];
    const float* C  = (const float*)d_in[5];

    float* out      = (float*)d_out;
    float* out_anti = out;                                 // [B, D]
    float* out_sym  = out + (size_t)B_TOT * D_DIM;         // [B, D]
    float* out_scal = out + 2 * (size_t)B_TOT * D_DIM;     // [B, 1]

    e8_bracket_kernel<<<BLOCKS, THREADS, 0, stream>>>(
        v1, v2, I, J, K, C, out_anti, out_sym, out_scal);
}